// ODEINDLayer_52845277610568
// MI455X (gfx1250) — compile-verified
//
#include <hip/hip_runtime.h>

// Problem constants (from reference): B,S,E,D,O = 512,512,4,2,3 ; K = D*O = 6
namespace {
constexpr int Bn = 512;
constexpr int Sn = 512;
constexpr int En = 4;
constexpr int Kn = 6;
}

typedef __attribute__((ext_vector_type(2))) float v2f;
typedef __attribute__((ext_vector_type(8))) float v8f;

// ---------------------------------------------------------------------------
// Phase 1: per (b,s) build A0 = c^T c (+ regularization) and beta = c^T rhs
// using one V_WMMA_F32_16X16X4_F32 per wave.
//   A operand (16x4): A[m][k] = coef[b,s,e=k,m]   (m<6, k<4)
//   B operand (4x16): B[k][n] = coef[b,s,e=k,n] for n<6 ; B[k][6] = rhs_eq[b,s,k]
//   => D[0:6,0:6] = Gram, D[0:6,6] = beta, one matrix op per grid point.
// ---------------------------------------------------------------------------
__global__ __launch_bounds__(256) void ode_build_kernel(
    const float* __restrict__ coef, const float* __restrict__ rhs_eq,
    const float* __restrict__ rhs_init, const float* __restrict__ steps,
    float* __restrict__ wsA, float* __restrict__ wsBeta)
{
  const int lane = threadIdx.x & 31;
  const int wave = blockIdx.x * (blockDim.x >> 5) + (threadIdx.x >> 5);
  const int b = wave / Sn;
  const int s = wave - b * Sn;
  const long base = (long)b * Sn + s;
  const float* cptr = coef + base * (En * Kn);

  const int m  = lane & 15;        // row (A) / col (B,C,D) within half-wave
  const int kh = (lane >> 4) << 1; // K-pair: 0 for lanes 0-15, 2 for lanes 16-31

  v2f aop; aop[0] = 0.f; aop[1] = 0.f;
  v2f bop; bop[0] = 0.f; bop[1] = 0.f;
  if (m < Kn) {
    const float c0 = cptr[(kh + 0) * Kn + m];
    const float c1 = cptr[(kh + 1) * Kn + m];
    aop[0] = c0; aop[1] = c1;   // A[m][kh], A[m][kh+1]
    bop[0] = c0; bop[1] = c1;   // B[kh][m], B[kh+1][m] (Gram is symmetric)
  } else if (m == Kn) {         // column 6 of B carries rhs_equation
    const float* rptr = rhs_eq + base * En;
    bop[0] = rptr[kh + 0];
    bop[1] = rptr[kh + 1];
  }

  v8f cacc = {};
  v8f dres = __builtin_amdgcn_wmma_f32_16x16x4_f32(
      false, aop, false, bop, (short)0, cacc, false, false);

  // D layout: lanes 0-15 hold rows 0..7 (VGPR r = row r) of column `lane`.
  if (lane < 16) {
    const int j = lane;
    if (j < Kn) {
      // ---- regularization block D0 (3x3), closed-form constants ----
      const float FtF[3][3] = {{1.f, 1.f, 0.5f}, {1.f, 2.f, 1.5f}, {0.5f, 1.5f, 1.25f}};
      const float SG[3][3]  = {{1.f, -1.f, 1.f}, {-1.f, 1.f, -1.f}, {1.f, -1.f, 1.f}};
      float D0[3][3];
#pragma unroll
      for (int a = 0; a < 3; ++a)
#pragma unroll
        for (int c = 0; c < 3; ++c) D0[a][c] = 0.f;
      const float* stp = steps + (size_t)b * (Sn - 1);
      if (s <= Sn - 2) {
        const float h = stp[s];
        float hp[5]; hp[0] = 1.f;
#pragma unroll
        for (int p = 1; p < 5; ++p) hp[p] = hp[p - 1] * h;
#pragma unroll
        for (int a = 0; a < 3; ++a)
#pragma unroll
          for (int c = 0; c < 3; ++c) D0[a][c] += hp[a + c] * FtF[a][c];
        D0[0][0] += 1.f;          // e^T e diag = {1, h^2, 0}
        D0[1][1] += h * h;
      }
      if (s >= 1) {
        const float g = stp[s - 1];
        float gp[5]; gp[0] = 1.f;
#pragma unroll
        for (int p = 1; p < 5; ++p) gp[p] = gp[p - 1] * g;
#pragma unroll
        for (int a = 0; a < 3; ++a)
#pragma unroll
          for (int c = 0; c < 3; ++c) D0[a][c] += gp[a + c] * FtF[a][c] * SG[a][c];
        D0[0][0] += 1.f;
        D0[1][1] += g * g;
      }
      if (s <= Sn - 3) D0[1][1] += 1.f;   // p26 = steps2^0 = 1
      if (s >= 2)      D0[1][1] += 1.f;
      if (s >= 1 && s <= Sn - 2) {        // p24 = steps2^2
        const float t = stp[s - 1] + stp[s];
        D0[2][2] += t * t;
      }
      float* Ao = wsA + base * 36;
#pragma unroll
      for (int i = 0; i < 6; ++i) {
        float v = dres[i];
        if ((i / 3) == (j / 3)) v += D0[i % 3][j % 3];          // blockdiag(D0)
        if (s < 2 && i == j && (i % 3) < 2) v += 1.f;           // init mask diag
        Ao[i * 6 + j] = v;
      }
    } else if (j == Kn) {
      // ---- beta column (+ rhs_init injection for first two steps) ----
      float* bo = wsBeta + base * 6;
#pragma unroll
      for (int i = 0; i < 6; ++i) {
        float v = dres[i];
        if (s < 2) {
          const int dd = i / 3, oo = i % 3;
          if (oo < 2) v += rhs_init[(((size_t)b * 2 + s) * 2 + dd) * 2 + oo];
        }
        bo[i] = v;
      }
    }
  }
}

// ---------------------------------------------------------------------------
// Phase 2: per-batch sequential block-Cholesky factorization scan.
// One thread per batch chain; all 6x6 state fully unrolled into VGPRs.
// L overwrites the A0 region in place.  B2 is the constant diag{-1 at k=1,4}.
// ---------------------------------------------------------------------------
__global__ __launch_bounds__(128) void ode_factor_kernel(
    const float* __restrict__ steps, float* __restrict__ wsL,
    float* __restrict__ wsT1, float* __restrict__ wsC2)
{
  const int b = blockIdx.x * blockDim.x + threadIdx.x;
  if (b >= Bn) return;
  const float* stp = steps + (size_t)b * (Sn - 1);

  float L1[36], L2[36], T1p[36];
#pragma unroll
  for (int i = 0; i < 36; ++i) {
    const float idv = (i % 7 == 0) ? 1.f : 0.f;
    L1[i] = idv; L2[i] = idv; T1p[i] = 0.f;
  }

  for (int s = 0; s < Sn; ++s) {
    const size_t base = ((size_t)b * Sn + s) * 36;
    if (s + 1 < Sn) __builtin_prefetch(&wsL[base + 36], 0, 0);

    float A[36];
#pragma unroll
    for (int i = 0; i < 36; ++i) A[i] = wsL[base + i];

    // ---- C2 = b2 @ L2^{-T} (right-solve; b2 diag{-1 at 1,4} when s>=2) ----
    float C2[36];
#pragma unroll
    for (int j = 0; j < 6; ++j) {
#pragma unroll
      for (int i = 0; i < 6; ++i) {
        float x = (s >= 2 && i == j && (i % 3) == 1) ? -1.f : 0.f;
#pragma unroll
        for (int mm = 0; mm < 6; ++mm)
          if (mm < j) x -= C2[i * 6 + mm] * L2[j * 6 + mm];
        C2[i * 6 + j] = x / L2[j * 6 + j];
      }
    }

    // ---- b1 = blockdiag(D1(steps[s-1])); T1 = (b1 - C2 T1p^T) @ L1^{-T} ----
    float D1m[3][3];
    {
      const float Mm[3][3] = {{2.f, 1.f, 0.5f}, {-1.f, 2.f, 1.f}, {0.5f, -1.f, 0.f}};
#pragma unroll
      for (int a = 0; a < 3; ++a)
#pragma unroll
        for (int c = 0; c < 3; ++c) D1m[a][c] = 0.f;
      if (s >= 1) {
        const float g = stp[s - 1];
        float gp[5]; gp[0] = 1.f;
#pragma unroll
        for (int p = 1; p < 5; ++p) gp[p] = gp[p - 1] * g;
#pragma unroll
        for (int a = 0; a < 3; ++a)
#pragma unroll
          for (int c = 0; c < 3; ++c) D1m[a][c] = -(gp[a + c] * Mm[a][c]);
        if (s <= Sn - 2) {                 // p25 correction
          const float t = g + stp[s];
          D1m[2][1] += t;
          D1m[1][2] -= t;
        }
      }
    }
    float T1[36];
#pragma unroll
    for (int i = 0; i < 6; ++i)
#pragma unroll
      for (int j = 0; j < 6; ++j) {
        float x = ((i / 3) == (j / 3)) ? D1m[i % 3][j % 3] : 0.f;
#pragma unroll
        for (int mm = 0; mm < 6; ++mm) x -= C2[i * 6 + mm] * T1p[j * 6 + mm];
        T1[i * 6 + j] = x;
      }
#pragma unroll
    for (int j = 0; j < 6; ++j) {        // in-place right-solve with L1
#pragma unroll
      for (int i = 0; i < 6; ++i) {
        float x = T1[i * 6 + j];
#pragma unroll
        for (int mm = 0; mm < 6; ++mm)
          if (mm < j) x -= T1[i * 6 + mm] * L1[j * 6 + mm];
        T1[i * 6 + j] = x / L1[j * 6 + j];
      }
    }

    // ---- Schur complement: A -= C2 C2^T + T1 T1^T ----
#pragma unroll
    for (int i = 0; i < 6; ++i)
#pragma unroll
      for (int j = 0; j < 6; ++j) {
        float x = A[i * 6 + j];
#pragma unroll
        for (int mm = 0; mm < 6; ++mm)
          x -= C2[i * 6 + mm] * C2[j * 6 + mm] + T1[i * 6 + mm] * T1[j * 6 + mm];
        A[i * 6 + j] = x;
      }

    // ---- 6x6 Cholesky (lower; zero upper) ----
    float L[36];
#pragma unroll
    for (int j = 0; j < 6; ++j) {
      float dv = A[j * 6 + j];
#pragma unroll
      for (int mm = 0; mm < 6; ++mm)
        if (mm < j) dv -= L[j * 6 + mm] * L[j * 6 + mm];
      const float dj = sqrtf(dv);
      L[j * 6 + j] = dj;
      const float inv = 1.f / dj;
#pragma unroll
      for (int i = 0; i < 6; ++i) {
        if (i < j) L[i * 6 + j] = 0.f;
        else if (i > j) {
          float x = A[i * 6 + j];
#pragma unroll
          for (int mm = 0; mm < 6; ++mm)
            if (mm < j) x -= L[i * 6 + mm] * L[j * 6 + mm];
          L[i * 6 + j] = x * inv;
        }
      }
    }

    // ---- store outputs, shift carry ----
#pragma unroll
    for (int i = 0; i < 36; ++i) {
      wsL[base + i]  = L[i];
      wsT1[base + i] = T1[i];
      wsC2[base + i] = C2[i];
      L2[i] = L1[i];
      L1[i] = L[i];
      T1p[i] = T1[i];
    }
  }
}

// ---------------------------------------------------------------------------
// Phase 3: forward substitution  y = L^{-1}(beta - T1 y1 - C2 y2)  (in place).
// ---------------------------------------------------------------------------
__global__ __launch_bounds__(128) void ode_fwd_kernel(
    const float* __restrict__ wsL, const float* __restrict__ wsT1,
    const float* __restrict__ wsC2, float* __restrict__ wsY)
{
  const int b = blockIdx.x * blockDim.x + threadIdx.x;
  if (b >= Bn) return;
  float y1[6], y2[6];
#pragma unroll
  for (int i = 0; i < 6; ++i) { y1[i] = 0.f; y2[i] = 0.f; }
  for (int s = 0; s < Sn; ++s) {
    const size_t b36 = ((size_t)b * Sn + s) * 36;
    const size_t b6  = ((size_t)b * Sn + s) * 6;
    float r[6];
#pragma unroll
    for (int i = 0; i < 6; ++i) {
      float acc = wsY[b6 + i];
#pragma unroll
      for (int mm = 0; mm < 6; ++mm)
        acc -= wsT1[b36 + i * 6 + mm] * y1[mm] + wsC2[b36 + i * 6 + mm] * y2[mm];
      r[i] = acc;
    }
    float y[6];
#pragma unroll
    for (int i = 0; i < 6; ++i) {
      float v = r[i];
#pragma unroll
      for (int mm = 0; mm < 6; ++mm)
        if (mm < i) v -= wsL[b36 + i * 6 + mm] * y[mm];
      y[i] = v / wsL[b36 + i * 6 + i];
    }
#pragma unroll
    for (int i = 0; i < 6; ++i) {
      wsY[b6 + i] = y[i];
      y2[i] = y1[i];
      y1[i] = y[i];
    }
  }
}

// ---------------------------------------------------------------------------
// Phase 4: backward substitution  x = L^{-T}(y - T1[s+1]^T x1 - C2[s+2]^T x2).
// Writes the final (B,S,D,O) output.
// ---------------------------------------------------------------------------
__global__ __launch_bounds__(128) void ode_bwd_kernel(
    const float* __restrict__ wsL, const float* __restrict__ wsT1,
    const float* __restrict__ wsC2, const float* __restrict__ wsY,
    float* __restrict__ out)
{
  const int b = blockIdx.x * blockDim.x + threadIdx.x;
  if (b >= Bn) return;
  float x1[6], x2[6];
#pragma unroll
  for (int i = 0; i < 6; ++i) { x1[i] = 0.f; x2[i] = 0.f; }
  for (int s = Sn - 1; s >= 0; --s) {
    const size_t b36 = ((size_t)b * Sn + s) * 36;
    const size_t b6  = ((size_t)b * Sn + s) * 6;
    float r[6];
#pragma unroll
    for (int i = 0; i < 6; ++i) r[i] = wsY[b6 + i];
    if (s + 1 < Sn) {
      const size_t n1 = b36 + 36;
#pragma unroll
      for (int i = 0; i < 6; ++i)
#pragma unroll
        for (int mm = 0; mm < 6; ++mm)
          r[i] -= wsT1[n1 + mm * 6 + i] * x1[mm];
    }
    if (s + 2 < Sn) {
      const size_t n2 = b36 + 72;
#pragma unroll
      for (int i = 0; i < 6; ++i)
#pragma unroll
        for (int mm = 0; mm < 6; ++mm)
          r[i] -= wsC2[n2 + mm * 6 + i] * x2[mm];
    }
    float x[6];
#pragma unroll
    for (int ii = 5; ii >= 0; --ii) {
      float v = r[ii];
#pragma unroll
      for (int mm = 0; mm < 6; ++mm)
        if (mm > ii) v -= wsL[b36 + mm * 6 + ii] * x[mm];
      x[ii] = v / wsL[b36 + ii * 6 + ii];
    }
#pragma unroll
    for (int i = 0; i < 6; ++i) {
      out[b6 + i] = x[i];
      x2[i] = x1[i];
      x1[i] = x[i];
    }
  }
}

extern "C" void kernel_launch(void* const* d_in, const int* in_sizes, int n_in,
                              void* d_out, int out_size, void* d_ws, size_t ws_size,
                              hipStream_t stream) {
  const float* coef     = (const float*)d_in[0]; // (B,S,E,D,O)
  const float* rhs_eq   = (const float*)d_in[1]; // (B,S,E)
  const float* rhs_init = (const float*)d_in[2]; // (B,2,D,2)
  const float* steps    = (const float*)d_in[3]; // (B,S-1)

  float* ws = (float*)d_ws;
  const size_t N36 = (size_t)Bn * Sn * 36;
  float* wsL  = ws;            // A0 built here, then L overwrites in place
  float* wsT1 = ws + N36;
  float* wsC2 = ws + 2 * N36;
  float* wsY  = ws + 3 * N36;  // beta built here, then y overwrites in place

  // Phase 1: one wave (one WMMA) per (b,s); 8 waves per 256-thread block.
  ode_build_kernel<<<dim3((Bn * Sn) / 8), dim3(256), 0, stream>>>(
      coef, rhs_eq, rhs_init, steps, wsL, wsY);
  // Phases 2-4: one thread per batch chain (sequential recurrences).
  ode_factor_kernel<<<dim3(Bn / 128), dim3(128), 0, stream>>>(steps, wsL, wsT1, wsC2);
  ode_fwd_kernel<<<dim3(Bn / 128), dim3(128), 0, stream>>>(wsL, wsT1, wsC2, wsY);
  ode_bwd_kernel<<<dim3(Bn / 128), dim3(128), 0, stream>>>(wsL, wsT1, wsC2, wsY,
                                                           (float*)d_out);
}